// GraphormerGraphNodeFeature_69930657513572
// MI455X (gfx1250) — compile-verified
//
#include <hip/hip_runtime.h>

// Problem constants (from reference)
static constexpr int kG     = 64;
static constexpr int kNodes = 511;
static constexpr int kN     = 512;   // NODES + 1
static constexpr int kH     = 768;
static constexpr int kE     = 4096;

typedef __attribute__((ext_vector_type(2))) float v2f;
typedef __attribute__((ext_vector_type(8))) float v8f;

// ---------------------------------------------------------------------------
// Kernel 1: node type table T[g,d] in [0,5): 0 = graph token, 1+t = node_emb[t]
// ---------------------------------------------------------------------------
__global__ void k_types(const int* __restrict__ nodes, int* __restrict__ T) {
  int idx = blockIdx.x * blockDim.x + threadIdx.x;
  if (idx >= kG * kN) return;
  int g = idx / kN, d = idx % kN;
  T[idx] = (d == 0) ? 0 : (1 + nodes[g * kNodes + (d - 1)]);
}

// ---------------------------------------------------------------------------
// Kernel 2: per-graph incoming-edge type histogram C[g,d,5] via LDS atomics.
// One block per graph; 512*5 u32 counters = 10 KB LDS (<< 320 KB/WGP).
// ---------------------------------------------------------------------------
__global__ void k_hist(const int* __restrict__ edges, const int* __restrict__ T,
                       unsigned* __restrict__ C) {
  __shared__ unsigned cnt[kN * 5];
  const int g = blockIdx.x;
  for (int i = threadIdx.x; i < kN * 5; i += blockDim.x) cnt[i] = 0u;
  __syncthreads();
  const int* src = edges + (size_t)g * 2 * kE;
  const int* dst = src + kE;
  const int* Tg  = T + g * kN;
  for (int e = threadIdx.x; e < kE; e += blockDim.x) {
    int s = src[e];
    int d = dst[e];
    int t = Tg[s];
    atomicAdd(&cnt[d * 5 + t], 1u);
  }
  __syncthreads();
  unsigned* Cg = C + (size_t)g * kN * 5;
  for (int i = threadIdx.x; i < kN * 5; i += blockDim.x) Cg[i] = cnt[i];
}

// ---------------------------------------------------------------------------
// Kernel 3: WMMA fp32 precompute of the rank-5 projection tables.
//   L[k,h] = emb5[k,:] . W_l[h,:]   (k<5)     R[k,h] = emb5[k,:] . W_r[h,:]
// emb5 row 0 = graph_emb, rows 1..4 = node_emb[0..3], rows 5..15 = 0 (pad).
// D = A(16x4) x B(4x16) + C accumulated over K=768 (192 x V_WMMA_F32_16X16X4_F32).
// One wave per 16-wide N-tile: 2 mats * 48 tiles = 96 waves. EXEC all-ones
// at every WMMA (mask-by-multiply instead of branches for the M>=5 pad).
// ---------------------------------------------------------------------------
__global__ void k_lr(const float* __restrict__ node_emb,
                     const float* __restrict__ graph_emb,
                     const float* __restrict__ Wl,
                     const float* __restrict__ Wr,
                     float* __restrict__ LR) {
  const int mat   = blockIdx.x / 48;     // 0 -> L (W_l), 1 -> R (W_r)
  const int ntile = blockIdx.x % 48;
  const float* W  = mat ? Wr : Wl;

  const int lane = threadIdx.x;          // wave32: 0..31
  const int half = lane >> 4;            // K-pair group
  const int lm   = lane & 15;            // M for A, N for B/D

  // A row (emb5 row lm), zero-masked for padded rows lm>=5
  int nr = lm - 1; if (nr < 0) nr = 0; if (nr > 3) nr = 3;
  const float* arow  = (lm == 0) ? graph_emb : (node_emb + nr * kH);
  const float  amask = (lm < 5) ? 1.0f : 0.0f;

  // B[k,n] = W[nglob, k]  (row of W, contiguous in k)
  const int    nglob = ntile * 16 + lm;
  const float* brow  = W + (size_t)nglob * kH;

  v8f acc = {};
  for (int k = 0; k < kH; k += 4) {
    const int ka = k + half * 2;
    v2f a, b;
    a.x = arow[ka]     * amask;
    a.y = arow[ka + 1] * amask;
    b.x = brow[ka];
    b.y = brow[ka + 1];
    acc = __builtin_amdgcn_wmma_f32_16x16x4_f32(
        /*neg_a=*/false, a, /*neg_b=*/false, b,
        /*c_mod=*/(short)0, acc, /*reuse_a=*/false, /*reuse_b=*/false);
  }

  // D layout: VGPR v holds M = v + 8*half, N = lane&15. Keep M<5 only.
  float* dstp = LR + (size_t)mat * 5 * kH;
#pragma unroll
  for (int v = 0; v < 8; ++v) {
    int m = v + 8 * half;
    if (m < 5) dstp[m * kH + nglob] = acc[v];
  }
}

// ---------------------------------------------------------------------------
// Kernel 4: streaming writer (HBM-bound, ~201 MB stores).
// One wave per (g,d) row; lane writes 6 float4 chunks of each output row.
//   out0[row] = emb5[t]
//   out1[row] = sum_k (C_k/max(deg,1)) * L[k] + b_l + R[t]
// ---------------------------------------------------------------------------
__global__ void k_out(const float* __restrict__ node_emb,
                      const float* __restrict__ graph_emb,
                      const float* __restrict__ b_l,
                      const int* __restrict__ T,
                      const unsigned* __restrict__ C,
                      const float* __restrict__ LR,
                      float* __restrict__ out0,
                      float* __restrict__ out1) {
  const int wave = (blockIdx.x * blockDim.x + threadIdx.x) >> 5;
  const int lane = threadIdx.x & 31;
  if (wave >= kG * kN) return;

  const int t = T[wave];
  const unsigned* c = C + (size_t)wave * 5;
  const float c0 = (float)c[0], c1 = (float)c[1], c2 = (float)c[2],
              c3 = (float)c[3], c4 = (float)c[4];
  const float inv = 1.0f / fmaxf(c0 + c1 + c2 + c3 + c4, 1.0f);
  const float w0 = c0 * inv, w1 = c1 * inv, w2 = c2 * inv,
              w3 = c3 * inv, w4 = c4 * inv;

  const float* L0 = LR;
  const float* L1 = LR + 1 * kH;
  const float* L2 = LR + 2 * kH;
  const float* L3 = LR + 3 * kH;
  const float* L4 = LR + 4 * kH;
  const float* R  = LR + 5 * kH + (size_t)t * kH;
  const float* xr = (t == 0) ? graph_emb : (node_emb + (size_t)(t - 1) * kH);

  float* o0 = out0 + (size_t)wave * kH;
  float* o1 = out1 + (size_t)wave * kH;

#pragma unroll
  for (int w = 0; w < 6; ++w) {
    const int h = (lane + w * 32) * 4;
    const float4 e  = *(const float4*)(xr + h);
    *(float4*)(o0 + h) = e;

    const float4 l0 = *(const float4*)(L0 + h);
    const float4 l1 = *(const float4*)(L1 + h);
    const float4 l2 = *(const float4*)(L2 + h);
    const float4 l3 = *(const float4*)(L3 + h);
    const float4 l4 = *(const float4*)(L4 + h);
    const float4 bb = *(const float4*)(b_l + h);
    const float4 rr = *(const float4*)(R + h);

    float4 r;
    r.x = fmaf(w0, l0.x, fmaf(w1, l1.x, fmaf(w2, l2.x, fmaf(w3, l3.x, fmaf(w4, l4.x, bb.x + rr.x)))));
    r.y = fmaf(w0, l0.y, fmaf(w1, l1.y, fmaf(w2, l2.y, fmaf(w3, l3.y, fmaf(w4, l4.y, bb.y + rr.y)))));
    r.z = fmaf(w0, l0.z, fmaf(w1, l1.z, fmaf(w2, l2.z, fmaf(w3, l3.z, fmaf(w4, l4.z, bb.z + rr.z)))));
    r.w = fmaf(w0, l0.w, fmaf(w1, l1.w, fmaf(w2, l2.w, fmaf(w3, l3.w, fmaf(w4, l4.w, bb.w + rr.w)))));
    *(float4*)(o1 + h) = r;
  }
}

// ---------------------------------------------------------------------------
extern "C" void kernel_launch(void* const* d_in, const int* in_sizes, int n_in,
                              void* d_out, int out_size, void* d_ws, size_t ws_size,
                              hipStream_t stream) {
  const int*   input_nodes = (const int*)d_in[0];
  const int*   input_edges = (const int*)d_in[1];
  const float* node_emb    = (const float*)d_in[2];
  const float* graph_emb   = (const float*)d_in[3];
  const float* W_l         = (const float*)d_in[4];
  const float* b_l         = (const float*)d_in[5];
  const float* W_r         = (const float*)d_in[6];

  float* out0 = (float*)d_out;                       // graph_node_feature
  float* out1 = out0 + (size_t)kG * kN * kH;         // graph_edge_feature

  // Workspace layout (<= 818 KB)
  char*     ws = (char*)d_ws;
  int*      T  = (int*)ws;                                   // 64*512*4      = 128 KB
  unsigned* C  = (unsigned*)(ws + (size_t)kG * kN * 4);      // 64*512*5*4    = 640 KB
  float*    LR = (float*)(ws + (size_t)kG * kN * 4
                             + (size_t)kG * kN * 5 * 4);     // 2*5*768*4     =  30 KB

  k_types<<<(kG * kN + 255) / 256, 256, 0, stream>>>(input_nodes, T);
  k_hist <<<kG, 256, 0, stream>>>(input_edges, T, C);
  k_lr   <<<96, 32, 0, stream>>>(node_emb, graph_emb, W_l, W_r, LR);
  k_out  <<<(kG * kN) / 8, 256, 0, stream>>>(node_emb, graph_emb, b_l, T, C, LR,
                                             out0, out1);
}